// DepthSeparableConv2d_conv3_3_37572373905420
// MI455X (gfx1250) — compile-verified
//
#include <hip/hip_runtime.h>
#include <hip/hip_bf16.h>

#define EPSV      1e-5f
#define DW_THRESH 4.0f
#define PW_THRESH 1e-3f

#define NB   32
#define NC   256
#define NO   256
#define HH   56
#define WW   56
#define HW   (HH * WW)      // 3136
#define HALO 58             // 56 + 2
#define WSTR 258            // LDS weight row stride (even -> 8B aligned pairs)

typedef __attribute__((ext_vector_type(2))) float v2f;
typedef __attribute__((ext_vector_type(8))) float v8f;

// ---------------------------------------------------------------------------
// Kernel 1: depthwise 3x3 conv + bias + BN1 + ReLU + per-plane max-abs cut.
// One workgroup per (b,c) plane so the plane max is a local reduction.
// Output layout is K-pair interleaved for kernel 2's WMMA B-loads:
//   y2[b][c>>1][pix][c&1]  (so channels {2k,2k+1} of one pixel are adjacent)
// ---------------------------------------------------------------------------
__global__ __launch_bounds__(256) void dw_bn_relu_cut_kernel(
    const float* __restrict__ x,   const float* __restrict__ dw_w,
    const float* __restrict__ dw_b,
    const float* __restrict__ g1,  const float* __restrict__ b1,
    const float* __restrict__ m1,  const float* __restrict__ v1,
    float* __restrict__ y2)
{
    __shared__ float tile[HALO * HALO];   // 58x58 halo tile (13.4 KB)
    __shared__ unsigned redmax;

    const int bc  = blockIdx.x;           // b*NC + c
    const int b   = bc >> 8;
    const int c   = bc & (NC - 1);
    const int tid = threadIdx.x;
    const float* __restrict__ xp = x + (size_t)bc * HW;

    // Load halo tile with zero padding.
    for (int idx = tid; idx < HALO * HALO; idx += 256) {
        const int r  = idx / HALO, q = idx % HALO;
        const int gh = r - 1,      gw = q - 1;
        float v = 0.0f;
        if ((unsigned)gh < (unsigned)HH && (unsigned)gw < (unsigned)WW)
            v = xp[gh * WW + gw];
        tile[idx] = v;
    }
    if (tid == 0) redmax = 0u;
    __syncthreads();

    // Per-channel constants.
    const float* wp = dw_w + c * 9;
    const float w0 = wp[0], w1 = wp[1], w2 = wp[2];
    const float w3 = wp[3], w4 = wp[4], w5 = wp[5];
    const float w6 = wp[6], w7 = wp[7], w8 = wp[8];
    const float bias  = dw_b[c];
    const float inv1  = g1[c] * rsqrtf(v1[c] + EPSV);
    const float sh1   = b1[c] - m1[c] * inv1;

    float vals[13];
    float lmax = 0.0f;
    int cnt = 0;
    for (int p = tid; p < HW; p += 256, ++cnt) {
        const int h = p / WW, w = p % WW;
        const float* t0 = &tile[h * HALO + w];    // top-left of 3x3 window
        float s = t0[0]          * w0 + t0[1]            * w1 + t0[2]            * w2
                + t0[HALO]       * w3 + t0[HALO + 1]     * w4 + t0[HALO + 2]     * w5
                + t0[2 * HALO]   * w6 + t0[2 * HALO + 1] * w7 + t0[2 * HALO + 2] * w8;
        float z = fmaxf((s + bias) * inv1 + sh1, 0.0f);   // BN1 + ReLU
        vals[cnt] = z;
        lmax = fmaxf(lmax, z);
    }
    // Post-ReLU values are >= 0, so float bit pattern order == numeric order.
    atomicMax(&redmax, __float_as_uint(lmax));
    __syncthreads();

    const bool keep = (__uint_as_float(redmax) >= DW_THRESH);
    // Pair-interleaved store: offset = b*NC*HW + (c>>1)*2*HW + 2*p + (c&1)
    float* __restrict__ yp = y2 + (size_t)b * NC * HW
                                + (size_t)(c >> 1) * (2 * HW) + (c & 1);
    cnt = 0;
    for (int p = tid; p < HW; p += 256, ++cnt)
        yp[2 * p] = keep ? vals[cnt] : 0.0f;
}

// ---------------------------------------------------------------------------
// Kernel 2: pointwise 1x1 conv as fp32 WMMA GEMM (16x16x4_f32) + bias + BN2
// + ReLU + per-(b,o)-plane cut. Grid = (b, o-group of 32). Each wave computes
// TWO 16x16 output tiles (o and o+16) that share every B load, and the pair-
// interleaved y layout makes each B operand a single global_load_b64.
// Each WG owns all 3136 pixels of one batch, so the plane max reduces locally.
// All epilogue loops are fully unrolled so accumulator accesses stay direct
// VGPR references (no m0/v_movrels indexed moves).
// ---------------------------------------------------------------------------
__global__ __launch_bounds__(256) void pw_wmma_bn_relu_cut_kernel(
    const float* __restrict__ y2,  const float* __restrict__ pw_w,
    const float* __restrict__ pw_b,
    const float* __restrict__ g2,  const float* __restrict__ b2,
    const float* __restrict__ m2,  const float* __restrict__ v2,
    float* __restrict__ zout)
{
    __shared__ float    wlds[32 * WSTR];  // [o_local][k], stride 258 (33 KB)
    __shared__ unsigned pmax[32];

    const int b    = blockIdx.x;
    const int o0   = blockIdx.y * 32;
    const int tid  = threadIdx.x;
    const int lane = tid & 31;
    const int wave = tid >> 5;            // 8 waves of 32

    // Stage weight tile for this o-group (32 x 256).
    for (int idx = tid; idx < 32 * 256; idx += 256) {
        const int o = idx >> 8, i = idx & 255;
        wlds[o * WSTR + i] = pw_w[(size_t)(o0 + o) * NC + i];
    }
    if (tid < 32) pmax[tid] = 0u;
    __syncthreads();

    // f32 WMMA lane mapping: lanes 0-15 carry K offsets {0,1}, lanes 16-31
    // carry K offsets {2,3}; lane%16 is the M (for A) / N (for B) index.
    const int hi = lane >> 4;             // 0 or 1
    const int ln = lane & 15;

    // Epilogue constants: accumulator row r of tile j maps to a fixed channel
    // o = o0 + 16*j + r + 8*hi (C/D layout: VGPR r, lane<16->M=r, else M=r+8).
    float inv2[16], add2[16];
#pragma unroll
    for (int j = 0; j < 2; ++j)
#pragma unroll
        for (int r = 0; r < 8; ++r) {
            const int o = o0 + 16 * j + r + hi * 8;
            const float iv = g2[o] * rsqrtf(v2[o] + EPSV);
            inv2[8 * j + r] = iv;
            add2[8 * j + r] = pw_b[o] * iv + (b2[o] - m2[o] * iv);
        }

    // A operands from LDS as aligned pairs: row ln (tile 0), row ln+16 (tile 1),
    // starting at column 2*hi; step k/2 pairs.
    const v2f* __restrict__ ap0 = (const v2f*)&wlds[ln * WSTR + 2 * hi];
    const v2f* __restrict__ ap1 = (const v2f*)&wlds[(ln + 16) * WSTR + 2 * hi];

    const float* __restrict__ yb = y2 + (size_t)b * NC * HW;
    float vmax[16];
#pragma unroll
    for (int r = 0; r < 16; ++r) vmax[r] = 0.0f;

    // 3136 pixels = 196 tiles of 16, distributed across the 8 waves.
    for (int t = wave; t < HW / 16; t += 8) {
        const int p0 = t * 16;
        v8f acc0 = {};
        v8f acc1 = {};
        // B operand base: rows (k+2*hi, k+2*hi+1) at pixel p0+ln are one
        // adjacent pair at float offset (k+2*hi)*HW + 2*(p0+ln).
        const v2f* __restrict__ bp =
            (const v2f*)(yb + (size_t)(2 * hi) * HW + 2 * (p0 + ln));
        for (int k = 0; k < NC; k += 4) {
            const v2f a0 = ap0[k >> 1];
            const v2f a1 = ap1[k >> 1];
            const v2f bm = bp[(size_t)k * (HW / 2)];   // global_load_b64
            acc0 = __builtin_amdgcn_wmma_f32_16x16x4_f32(
                false, a0, false, bm, (short)0, acc0, false, false);
            acc1 = __builtin_amdgcn_wmma_f32_16x16x4_f32(
                false, a1, false, bm, (short)0, acc1, false, false);
        }
        // Epilogue: bias + BN2 + ReLU, store, track per-o plane max.
        // Fully unrolled: acc subscripts must stay compile-time constant.
#pragma unroll
        for (int r = 0; r < 8; ++r) {
            float z0 = fmaxf(acc0[r] * inv2[r] + add2[r], 0.0f);
            float z1 = fmaxf(acc1[r] * inv2[8 + r] + add2[8 + r], 0.0f);
            vmax[r]     = fmaxf(vmax[r], z0);
            vmax[8 + r] = fmaxf(vmax[8 + r], z1);
            zout[(size_t)(b * NO + o0 + r + hi * 8) * HW + p0 + ln] = z0;
            zout[(size_t)(b * NO + o0 + 16 + r + hi * 8) * HW + p0 + ln] = z1;
        }
    }

#pragma unroll
    for (int r = 0; r < 8; ++r) {
        atomicMax(&pmax[r + hi * 8],      __float_as_uint(vmax[r]));
        atomicMax(&pmax[16 + r + hi * 8], __float_as_uint(vmax[8 + r]));
    }
    __threadfence_block();
    __syncthreads();

    // Zero whole planes whose post-ReLU max is below threshold.
    for (int o = 0; o < 32; ++o) {
        if (__uint_as_float(pmax[o]) < PW_THRESH) {
            float* __restrict__ zp = zout + (size_t)(b * NO + o0 + o) * HW;
            for (int p = tid; p < HW; p += 256) zp[p] = 0.0f;
        }
    }
}

// ---------------------------------------------------------------------------
extern "C" void kernel_launch(void* const* d_in, const int* in_sizes, int n_in,
                              void* d_out, int out_size, void* d_ws, size_t ws_size,
                              hipStream_t stream) {
    const float* x    = (const float*)d_in[0];
    const float* dw_w = (const float*)d_in[1];
    const float* dw_b = (const float*)d_in[2];
    const float* g1   = (const float*)d_in[3];
    const float* b1   = (const float*)d_in[4];
    const float* m1   = (const float*)d_in[5];
    const float* v1   = (const float*)d_in[6];
    const float* pw_w = (const float*)d_in[7];
    const float* pw_b = (const float*)d_in[8];
    const float* g2   = (const float*)d_in[9];
    const float* b2   = (const float*)d_in[10];
    const float* m2   = (const float*)d_in[11];
    const float* v2   = (const float*)d_in[12];

    float* y2   = (float*)d_ws;        // intermediate, pair-interleaved, 102.8 MB
    float* zout = (float*)d_out;

    dw_bn_relu_cut_kernel<<<NB * NC, 256, 0, stream>>>(
        x, dw_w, dw_b, g1, b1, m1, v1, y2);

    pw_wmma_bn_relu_cut_kernel<<<dim3(NB, NO / 32), 256, 0, stream>>>(
        y2, pw_w, pw_b, g2, b2, m2, v2, zout);

    (void)in_sizes; (void)n_in; (void)out_size; (void)ws_size;
}